// GCNLinkPredictor_18648747999234
// MI455X (gfx1250) — compile-verified
//
#include <hip/hip_runtime.h>
#include <hip/hip_bf16.h>

typedef __attribute__((ext_vector_type(2))) float v2f;
typedef __attribute__((ext_vector_type(8))) float v8f;

#define IN_C  128
#define HID_C 128
#define OUT_C 64

// ---------------------------------------------------------------------------
// Degree / norm kernels
// ---------------------------------------------------------------------------
__global__ void deg_init(float* __restrict__ deg, int n) {
  int i = blockIdx.x * blockDim.x + threadIdx.x;
  if (i < n) deg[i] = 1.0f;  // self-loop contribution
}

__global__ void deg_count(const int* __restrict__ dst, int E, float* __restrict__ deg) {
  int i = blockIdx.x * blockDim.x + threadIdx.x;
  if (i < E) atomicAdd(&deg[dst[i]], 1.0f);
}

__global__ void deg_finish(float* __restrict__ deg, int n) {
  int i = blockIdx.x * blockDim.x + threadIdx.x;
  if (i < n) deg[i] = rsqrtf(deg[i]);  // deg >= 1 always
}

// ---------------------------------------------------------------------------
// Generic fp32 WMMA GEMM:  C[M,N] = A[M,K] @ B[K,N]   (row-major)
// M % 16 == 0, N % 16 == 0, K % 4 == 0.  One wave -> one 16x16 C tile.
// A frag (16x4 f32): lane l<16 holds row m0+l, K = k,k+1 ; lanes 16-31 K = k+2,k+3
// B frag (4x16 f32): lane l<16 holds col n0+l, K = k,k+1 ; lanes 16-31 K = k+2,k+3
// C frag: VGPR i -> M = m0 + i + (half?8:0), N = n0 + (lane&15)
// ---------------------------------------------------------------------------
__global__ void wmma_gemm_f32(const float* __restrict__ A, const float* __restrict__ B,
                              float* __restrict__ C, int M, int N, int K) {
  int wave = (int)((blockIdx.x * (long long)blockDim.x + threadIdx.x) >> 5);
  int lane = threadIdx.x & 31;
  int ntiles = N >> 4;
  int total = (M >> 4) * ntiles;
  if (wave >= total) return;  // whole-wave uniform: EXEC stays all-ones
  int mt = wave / ntiles, nt = wave - mt * ntiles;
  int m0 = mt << 4, n0 = nt << 4;
  int half = lane >> 4;
  int l = lane & 15;

  const float* Arow = A + (size_t)(m0 + l) * K + 2 * half;
  const float* Bcol = B + (size_t)(2 * half) * N + n0 + l;

  v8f acc = {};
  for (int k = 0; k < K; k += 4) {
    v2f a, b;
    a.x = Arow[k];
    a.y = Arow[k + 1];
    b.x = Bcol[(size_t)k * N];
    b.y = Bcol[(size_t)(k + 1) * N];
    acc = __builtin_amdgcn_wmma_f32_16x16x4_f32(false, a, false, b, (short)0, acc,
                                                false, false);
  }

  float* Crow = C + (size_t)(m0 + 8 * half) * N + n0 + l;
#pragma unroll
  for (int i = 0; i < 8; ++i) Crow[(size_t)i * N] = acc[i];
}

// ---------------------------------------------------------------------------
// GCN aggregation: out = sym-norm scatter-add of h, then bias (+ optional relu)
// ---------------------------------------------------------------------------
__global__ void agg_self(const float* __restrict__ h, const float* __restrict__ dinv,
                         float* __restrict__ out, long long total, int C) {
  long long t = blockIdx.x * (long long)blockDim.x + threadIdx.x;
  if (t >= total) return;
  int v = (int)(t / C);
  float di = dinv[v];
  out[t] = h[t] * di * di;  // self-loop: norm = dinv[v]*dinv[v]
}

__global__ void agg_scatter(const float* __restrict__ h, const int* __restrict__ src,
                            const int* __restrict__ dst, const float* __restrict__ dinv,
                            float* __restrict__ out, int E, int C) {
  long long t = blockIdx.x * (long long)blockDim.x + threadIdx.x;
  int groups = C >> 2;  // float4 channel groups
  long long total = (long long)E * groups;
  if (t >= total) return;
  int e  = (int)(t / groups);
  int c4 = (int)(t - (long long)e * groups) << 2;
  int s = src[e], d = dst[e];
  float nrm = dinv[s] * dinv[d];
  float4 hv = *(const float4*)(h + (size_t)s * C + c4);
  float* o = out + (size_t)d * C + c4;
  atomicAdd(o + 0, hv.x * nrm);
  atomicAdd(o + 1, hv.y * nrm);
  atomicAdd(o + 2, hv.z * nrm);
  atomicAdd(o + 3, hv.w * nrm);
}

__global__ void bias_act(float* __restrict__ out, const float* __restrict__ b,
                         long long total, int C, int do_relu) {
  long long t = blockIdx.x * (long long)blockDim.x + threadIdx.x;
  if (t >= total) return;
  float v = out[t] + b[(int)(t % C)];
  out[t] = do_relu ? fmaxf(v, 0.0f) : v;
}

// ---------------------------------------------------------------------------
// MLP decode: per pair p, feat = [z[s], z[d]] (128), hid = relu(feat@Wa + ba),
// out[p] = hid . Wb + bb.  One wave -> 16 pairs.  Layer-A via WMMA f32
// (8 N-tiles x 32 K-steps), layer-B folded into a shfl_xor half-wave reduce.
// ---------------------------------------------------------------------------
__global__ void decode_mlp(const float* __restrict__ z,     // [n, OUT_C]
                           const int* __restrict__ sidx, const int* __restrict__ didx,
                           const float* __restrict__ Wa,    // [2*OUT_C, HID_C]
                           const float* __restrict__ ba,    // [HID_C]
                           const float* __restrict__ Wb,    // [HID_C]
                           const float* __restrict__ bb,    // [1]
                           float* __restrict__ out, int P) {
  int wave = (int)((blockIdx.x * (long long)blockDim.x + threadIdx.x) >> 5);
  int lane = threadIdx.x & 31;
  if (wave >= (P >> 4)) return;  // uniform per wave
  int m0 = wave << 4;
  int half = lane >> 4;
  int l = lane & 15;

  int row = m0 + l;
  int s = sidx[row], d = didx[row];
  const float* zs = z + (size_t)s * OUT_C;
  const float* zd = z + (size_t)d * OUT_C;

  v8f acc[8];
#pragma unroll
  for (int t = 0; t < 8; ++t) acc[t] = (v8f){};

  for (int k = 0; k < 2 * OUT_C; k += 4) {
    int ka = k + 2 * half;
    v2f a;
    a.x = (ka     < OUT_C) ? zs[ka]         : zd[ka - OUT_C];
    a.y = (ka + 1 < OUT_C) ? zs[ka + 1]     : zd[ka + 1 - OUT_C];
    const float* Bp = Wa + (size_t)ka * HID_C + l;
#pragma unroll
    for (int t = 0; t < 8; ++t) {
      v2f b;
      b.x = Bp[t * 16];
      b.y = Bp[HID_C + t * 16];
      acc[t] = __builtin_amdgcn_wmma_f32_16x16x4_f32(false, a, false, b, (short)0,
                                                     acc[t], false, false);
    }
  }

  // hid = relu(acc + ba[n]); partial[i] = sum over this lane's column of hid*Wb
  float part[8];
#pragma unroll
  for (int i = 0; i < 8; ++i) part[i] = 0.0f;
#pragma unroll
  for (int t = 0; t < 8; ++t) {
    int nidx = t * 16 + l;
    float bav = ba[nidx];
    float wbv = Wb[nidx];
#pragma unroll
    for (int i = 0; i < 8; ++i) {
      float hv = fmaxf(acc[t][i] + bav, 0.0f);
      part[i] += hv * wbv;
    }
  }
  // reduce across the 16 lanes of each half-wave (rows i / 8+i)
#pragma unroll
  for (int m = 1; m <= 8; m <<= 1) {
#pragma unroll
    for (int i = 0; i < 8; ++i) part[i] += __shfl_xor(part[i], m, 32);
  }
  if (l == 0) {
    float bbv = bb[0];
#pragma unroll
    for (int i = 0; i < 8; ++i) out[m0 + 8 * half + i] = part[i] + bbv;
  }
}

// ---------------------------------------------------------------------------
// Host launcher
// ---------------------------------------------------------------------------
extern "C" void kernel_launch(void* const* d_in, const int* in_sizes, int n_in,
                              void* d_out, int out_size, void* d_ws, size_t ws_size,
                              hipStream_t stream) {
  const float* x   = (const float*)d_in[0];
  const int* eidx  = (const int*)d_in[1];
  const int* pidx  = (const int*)d_in[2];
  const int* nidx  = (const int*)d_in[3];
  const float* W1  = (const float*)d_in[4];
  const float* b1  = (const float*)d_in[5];
  const float* W2  = (const float*)d_in[6];
  const float* b2  = (const float*)d_in[7];
  const float* Wa  = (const float*)d_in[8];
  const float* ba  = (const float*)d_in[9];
  const float* Wb  = (const float*)d_in[10];
  const float* bb  = (const float*)d_in[11];
  float* out = (float*)d_out;

  const int n = in_sizes[0] / IN_C;        // 50000
  const int E = in_sizes[1] / 2;           // 800000
  const int P = in_sizes[2] / 2;           // 200000
  const int* e_src = eidx;
  const int* e_dst = eidx + E;

  // workspace layout (fp32)
  float* ws   = (float*)d_ws;
  float* h1   = ws;                        // n*128 (reused as h2 [n*64] | z2 [n*64])
  float* z1   = h1 + (size_t)n * HID_C;    // n*128
  float* dinv = z1 + (size_t)n * HID_C;    // n
  float* h2   = h1;                        // alias: h1 dead after z1 is built
  float* z2   = h1 + (size_t)n * OUT_C;    // second half of the h1 block

  auto cdiv = [](long long a, long long b) { return (int)((a + b - 1) / b); };
  const int B = 256;

  // --- dinv = rsqrt(1 + indegree) --------------------------------------
  deg_init<<<cdiv(n, B), B, 0, stream>>>(dinv, n);
  deg_count<<<cdiv(E, B), B, 0, stream>>>(e_dst, E, dinv);
  deg_finish<<<cdiv(n, B), B, 0, stream>>>(dinv, n);

  // --- layer 1: h1 = x @ W1 ; z1 = relu(agg(h1) + b1) -------------------
  {
    long long waves = (long long)(n / 16) * (HID_C / 16);
    wmma_gemm_f32<<<cdiv(waves * 32, B), B, 0, stream>>>(x, W1, h1, n, HID_C, IN_C);
    long long tot = (long long)n * HID_C;
    agg_self<<<cdiv(tot, B), B, 0, stream>>>(h1, dinv, z1, tot, HID_C);
    agg_scatter<<<cdiv((long long)E * (HID_C / 4), B), B, 0, stream>>>(
        h1, e_src, e_dst, dinv, z1, E, HID_C);
    bias_act<<<cdiv(tot, B), B, 0, stream>>>(z1, b1, tot, HID_C, 1);
  }

  // --- layer 2: h2 = z1 @ W2 ; z2 = agg(h2) + b2 ------------------------
  {
    long long waves = (long long)(n / 16) * (OUT_C / 16);
    wmma_gemm_f32<<<cdiv(waves * 32, B), B, 0, stream>>>(z1, W2, h2, n, OUT_C, HID_C);
    long long tot = (long long)n * OUT_C;
    agg_self<<<cdiv(tot, B), B, 0, stream>>>(h2, dinv, z2, tot, OUT_C);
    agg_scatter<<<cdiv((long long)E * (OUT_C / 4), B), B, 0, stream>>>(
        h2, e_src, e_dst, dinv, z2, E, OUT_C);
    bias_act<<<cdiv(tot, B), B, 0, stream>>>(z2, b2, tot, OUT_C, 0);
  }

  // --- decode: pos then neg --------------------------------------------
  {
    long long waves = P / 16;
    decode_mlp<<<cdiv(waves * 32, B), B, 0, stream>>>(z2, pidx, pidx + P, Wa, ba, Wb,
                                                      bb, out, P);
    decode_mlp<<<cdiv(waves * 32, B), B, 0, stream>>>(z2, nidx, nidx + P, Wa, ba, Wb,
                                                      bb, out + P, P);
  }
}